// Query2Context_15539191677614
// MI455X (gfx1250) — compile-verified
//
#include <hip/hip_runtime.h>
#include <math.h>

// Problem constants (from reference setup_inputs)
#define TT 16384   // context length T
#define JJ 64      // query axis J
#define DD 1024    // feature dim d

typedef __attribute__((ext_vector_type(2))) float v2f;
typedef __attribute__((ext_vector_type(4))) float v4f;
typedef __attribute__((ext_vector_type(8))) float v8f;

// ---------------------------------------------------------------------------
// K1: m[t] = max_j s[t, j]   (s is [T, J] row-major, J = 64)
// Each thread reads its 256-byte row with 16x b128 loads.
// ---------------------------------------------------------------------------
__global__ void q2c_rowmax(const float* __restrict__ s, float* __restrict__ m) {
  int t = blockIdx.x * blockDim.x + threadIdx.x;
  if (t >= TT) return;
  const v4f* row = (const v4f*)(s + (size_t)t * JJ);
  float mx = -INFINITY;
#pragma unroll
  for (int i = 0; i < JJ / 4; ++i) {
    v4f v = row[i];
    mx = fmaxf(mx, fmaxf(fmaxf(v.x, v.y), fmaxf(v.z, v.w)));
  }
  m[t] = mx;
}

// ---------------------------------------------------------------------------
// K2: single-workgroup softmax over the 16384 row-maxima.
//   w[t] = exp(m[t] - M) / sum   ;  also zero-init ctx accumulator.
// ws layout: m = ws[0..T), w = ws[T..2T), ctx = ws[2T..2T+D)
// ---------------------------------------------------------------------------
__global__ void q2c_softmax(float* __restrict__ ws) {
  float* m   = ws;
  float* w   = ws + TT;
  float* ctx = ws + 2 * TT;
  __shared__ float red[1024];
  int tid = threadIdx.x;

  // global max
  float mx = -INFINITY;
  for (int i = tid; i < TT; i += 1024) mx = fmaxf(mx, m[i]);
  red[tid] = mx;
  __syncthreads();
  for (int off = 512; off > 0; off >>= 1) {
    if (tid < off) red[tid] = fmaxf(red[tid], red[tid + off]);
    __syncthreads();
  }
  float M = red[0];
  __syncthreads();

  // exp + sum
  float sum = 0.0f;
  for (int i = tid; i < TT; i += 1024) {
    float e = __expf(m[i] - M);
    w[i] = e;
    sum += e;
  }
  red[tid] = sum;
  __syncthreads();
  for (int off = 512; off > 0; off >>= 1) {
    if (tid < off) red[tid] += red[tid + off];
    __syncthreads();
  }
  float inv = 1.0f / red[0];
  __syncthreads();

  // normalize; zero ctx accumulator (blockDim.x == 1024 == DD)
  for (int i = tid; i < TT; i += 1024) w[i] *= inv;
  ctx[tid] = 0.0f;
}

// ---------------------------------------------------------------------------
// K3: ctx[k] = sum_t w[t] * h[t, k]  via V_WMMA_F32_16X16X4_F32.
//
// Per wave (32 lanes): one 16-column tile (k0..k0+15) and one T-split of
// 128 rows, 4 rows per WMMA (K = 4).
//   A (16x4 f32, 2 VGPRs/lane): row 0 = w[t..t+3], rows 1..15 = 0.
//     ISA layout: lanes 0-15 hold M=lane {K0,K1}; lanes 16-31 hold
//     M=lane-16 {K2,K3}. Lane 0 needs {w[t],w[t+1]}, lane 16 needs
//     {w[t+2],w[t+3]} == w[t+2*half .. +1]. Branch-free build: every lane
//     does one b64 load of w+t+2*half and masks with (col==0) via cndmask
//     -> no EXEC divergence, no per-iteration branch overhead.
//   B (4x16 f32, 2 VGPRs/lane): VGPR0 = row K0 (lanes 0-15) / K2
//     (lanes 16-31); VGPR1 = K1 / K3.  B element = h[(t+2*half+v)*D+k0+col].
//   C/D (16x16 f32, 8 VGPRs): row 0 of D lives in c[0] of lanes 0-15.
// Partial ctx accumulated with f32 atomics (128 adds/element, negligible).
// Memory-bound: the 67 MB stream of h dominates; f32 WMMA keeps reference
// precision at trivial FLOP cost.
// ---------------------------------------------------------------------------
#define NSPLIT 128
#define ROWS_PER_SPLIT (TT / NSPLIT)   // 128

__global__ void q2c_wsum_wmma(const float* __restrict__ h, float* __restrict__ ws) {
  const float* w = ws + TT;
  float* ctx     = ws + 2 * TT;

  int lane = threadIdx.x & 31;
  int wave = threadIdx.x >> 5;              // 0..7
  int tile = blockIdx.x * 8 + wave;         // 0..63 column tile
  int k0   = tile * 16;
  int t0   = blockIdx.y * ROWS_PER_SPLIT;   // blockIdx.y in [0, NSPLIT)

  int col  = lane & 15;
  int half = lane >> 4;
  bool row0 = (col == 0);

  // lane-invariant within iteration: base pointers
  const float* hp = h + (size_t)(t0 + 2 * half) * DD + k0 + col;  // row t+2*half
  const float* wp = w + t0 + 2 * half;

  v8f c = {};
#pragma unroll 8
  for (int i = 0; i < ROWS_PER_SPLIT / 4; ++i) {
    // A: one b64 load of the two weights this half-wave covers, masked to col 0
    v2f wv = *(const v2f*)(wp);
    v2f A;
    A.x = row0 ? wv.x : 0.0f;
    A.y = row0 ? wv.y : 0.0f;

    // B: rows t+2*half and t+2*half+1 of the 16-column tile
    v2f B;
    B.x = hp[0];
    B.y = hp[DD];

    c = __builtin_amdgcn_wmma_f32_16x16x4_f32(
        /*neg_a=*/false, A, /*neg_b=*/false, B,
        /*c_mod=*/(short)0, c, /*reuse_a=*/false, /*reuse_b=*/false);

    wp += 4;
    hp += 4 * DD;
  }

  // Row 0 of D = partial ctx for columns k0..k0+15 (lanes 0-15, c[0]).
  if (lane < 16) atomicAdd(&ctx[k0 + col], c[0]);
}

// ---------------------------------------------------------------------------
// K4: out[k, t] = ctx[k]   (out is [d, T] row-major, 67 MB)
// Nontemporal b128 stores: the output is write-once and read back by the
// host; keeping it out of L2 leaves room for h (67 MB) to stay resident
// across graph replays.
// ---------------------------------------------------------------------------
__global__ void q2c_bcast(const float* __restrict__ ws, float* __restrict__ out) {
  const float* ctx = ws + 2 * TT;
  int k = blockIdx.y;
  float v = ctx[k];
  v4f val = {v, v, v, v};
  size_t x = ((size_t)blockIdx.x * blockDim.x + threadIdx.x) * 4;
  __builtin_nontemporal_store(val, (v4f*)(out + (size_t)k * TT + x));
}

// ---------------------------------------------------------------------------
extern "C" void kernel_launch(void* const* d_in, const int* in_sizes, int n_in,
                              void* d_out, int out_size, void* d_ws, size_t ws_size,
                              hipStream_t stream) {
  const float* h = (const float*)d_in[0];   // [1, T, d] f32
  const float* s = (const float*)d_in[1];   // [T, J] f32
  float* out = (float*)d_out;               // [d, T] f32
  float* ws  = (float*)d_ws;                // m[T] | w[T] | ctx[D]

  // K1: row maxima
  q2c_rowmax<<<TT / 256, 256, 0, stream>>>(s, ws);

  // K2: softmax over T (single WG) + ctx zero-init
  q2c_softmax<<<1, 1024, 0, stream>>>(ws);

  // K3: WMMA weighted column-sum: 8 blocks x 8 waves = 64 column tiles,
  //     NSPLIT T-splits -> 8192 waves total
  q2c_wsum_wmma<<<dim3(8, NSPLIT), 256, 0, stream>>>(h, ws);

  // K4: broadcast ctx across T: grid (T/1024, d), float4 per thread
  q2c_bcast<<<dim3(TT / 1024, DD), 256, 0, stream>>>(ws, out);
}